// QuantumConv2D_27822798143891
// MI455X (gfx1250) — compile-verified
//
#include <hip/hip_runtime.h>
#include <math.h>

// out[n,i,j] = prod_{3x3 window} cos(x), out-of-range taps -> cos(lower-median of plane n)
// Derivation: RY-product state + CNOT chain => <Z_last> = prod_q cos(theta_q); ancilla cos(0)=1.

#define NPLANES   12        // B*C = 4*3
#define PLANE     4096      // 64*64
#define WDIM      64
#define THREADS   256       // 8 wave32 waves
#define PER_THR   16        // 4096 / 256
#define KRANK     2047u     // lower median: sorted[(4096-1)/2]

// ---- CDNA5 TDM probe: builtin exists on both toolchains but with different arity.
//      The gfx1250 TDM header ships only with the 6-arg (therock-10.0) toolchain.
#if __has_builtin(__builtin_amdgcn_tensor_load_to_lds)
#  define QC_TDM 1
#  if __has_include(<hip/amd_detail/amd_gfx1250_TDM.h>)
#    define QC_TDM_ARGS 6
#  else
#    define QC_TDM_ARGS 5
#  endif
#else
#  define QC_TDM 0
#endif

typedef uint32_t v4u __attribute__((ext_vector_type(4)));
typedef int      v8i __attribute__((ext_vector_type(8)));
typedef int      v4i __attribute__((ext_vector_type(4)));

__device__ __forceinline__ void wait_tensorcnt0() {
#if __has_builtin(__builtin_amdgcn_s_wait_tensorcnt)
    __builtin_amdgcn_s_wait_tensorcnt((short)0);
#else
    asm volatile("s_wait_tensorcnt 0x0" ::: "memory");
#endif
}

__device__ __forceinline__ void async_copy_b128(uint32_t lds_byte_addr, const float* gaddr) {
    // CDNA5: GLOBAL_LOAD_ASYNC_TO_LDS_B128 — per-lane 16B DMA global->LDS, tracked by ASYNCcnt
    asm volatile("global_load_async_to_lds_b128 %0, %1, off"
                 :: "v"(lds_byte_addr), "v"(gaddr)
                 : "memory");
}

__device__ __forceinline__ void wait_asynccnt0() {
#if __has_builtin(__builtin_amdgcn_s_wait_asynccnt)
    __builtin_amdgcn_s_wait_asynccnt(0);
#else
    asm volatile("s_wait_asynccnt 0x0" ::: "memory");
#endif
}

__global__ __launch_bounds__(THREADS)
void qconv_median_cosprod_kernel(const float* __restrict__ x, float* __restrict__ out) {
    __shared__ __align__(16) uint32_t uu[PLANE];  // raw x bits (TDM dest), then order-mapped uints
    __shared__ float    cb[PLANE];                // cos(x)
    __shared__ uint32_t hist[256];
    __shared__ uint32_t sPrefix;
    __shared__ uint32_t sRank;
    __shared__ float    sCosM;

    const int tid = threadIdx.x;
    const int n   = blockIdx.x;
    const float* src = x + (size_t)n * PLANE;
    const uint32_t ldsBase = (uint32_t)(uintptr_t)&uu[0];

    // ---- Phase 1: bulk copy plane (16 KB) global -> LDS ----
#if QC_TDM
    // Tensor Data Mover: one descriptor per block moves the whole 64x64 f32 tile.
    if (tid == 0) {
        const uint64_t ga = (uint64_t)(uintptr_t)src;
        v4u g0;
        g0.x = 1u;                                               // count=1 (valid user D#)
        g0.y = ldsBase;                                          // lds_addr (bytes)
        g0.z = (uint32_t)ga;                                     // global_addr[31:0]
        g0.w = (uint32_t)((ga >> 32) & 0x01FFFFFFu) | (2u << 30);// global_addr[56:32] | type=2
        v8i g1;
        g1[0] = (int)(2u << 16);        // data_size = 4B (bits 17:16)
        g1[1] = (int)(64u << 16);       // tensor_dim0 = 64   (bits 79:48)
        g1[2] = (int)(64u << 16);       // tensor_dim1 = 64   (bits 111:80)
        g1[3] = (int)(64u << 16);       // tile_dim0   = 64   (bits 127:112)
        g1[4] = 64;                     // tile_dim1   = 64   (bits 143:128); tile_dim2=0 -> 2D
        g1[5] = 64;                     // tensor_dim0_stride = 64 (bits 207:160)
        g1[6] = (int)(4096u << 16);     // tensor_dim1_stride = 4096 (bits 255:208), unused for 2D
        g1[7] = 0;
        v4i gz = {0, 0, 0, 0};
#  if QC_TDM_ARGS == 6
        v8i z8 = {0, 0, 0, 0, 0, 0, 0, 0};
        __builtin_amdgcn_tensor_load_to_lds(g0, g1, gz, gz, z8, 0);
#  else
        __builtin_amdgcn_tensor_load_to_lds(g0, g1, gz, gz, 0);
#  endif
        wait_tensorcnt0();              // TENSORcnt==0 -> tile resident in LDS
    }
    __syncthreads();
#else
    // Fallback: per-lane async B128 copies, tracked by ASYNCcnt.
#pragma unroll
    for (int k = 0; k < 4; ++k) {
        const int v4 = tid + THREADS * k;                  // float4 index 0..1023
        async_copy_b128(ldsBase + (uint32_t)v4 * 16u, src + v4 * 4);
    }
    wait_asynccnt0();
    __syncthreads();
#endif

    // ---- Phase 2: per-element cos + order-preserving uint map (each thread owns its slots) ----
#pragma unroll 1
    for (int k = 0; k < PER_THR; ++k) {
        const int idx = tid + THREADS * k;
        const uint32_t fb = uu[idx];
        cb[idx] = cosf(__uint_as_float(fb));
        uu[idx] = (fb & 0x80000000u) ? ~fb : (fb | 0x80000000u);
    }
    if (tid == 0) { sPrefix = 0u; sRank = KRANK; }
    __syncthreads();

    // ---- Phase 3: exact rank-select (lower median) via 4 x 8-bit radix passes in LDS ----
#pragma unroll 1
    for (int pass = 0; pass < 4; ++pass) {
        const int shift = 24 - pass * 8;
        const uint32_t mask = (pass == 0) ? 0u : (0xFFFFFFFFu << (shift + 8));
        hist[tid & 255] = 0u;
        __syncthreads();
        const uint32_t pfx = sPrefix;
#pragma unroll 1
        for (int k = 0; k < PER_THR; ++k) {
            const uint32_t m = uu[tid + THREADS * k];
            if ((m & mask) == pfx) atomicAdd(&hist[(m >> shift) & 255u], 1u);
        }
        __syncthreads();
        if (tid == 0) {
            uint32_t rank = sRank, acc = 0u, b = 0u;
            for (b = 0u; b < 256u; ++b) {
                const uint32_t h = hist[b];
                if (acc + h > rank) break;
                acc += h;
            }
            sRank   = rank - acc;
            sPrefix = pfx | (b << shift);
        }
        __syncthreads();
    }

    // ---- Phase 4: unmap median bits -> cos(median) for the padding taps ----
    if (tid == 0) {
        const uint32_t mu = sPrefix;
        const uint32_t fb = (mu & 0x80000000u) ? (mu & 0x7FFFFFFFu) : ~mu;
        sCosM = cosf(__uint_as_float(fb));
    }
    __syncthreads();
    const float cosm = sCosM;

    // ---- Phase 5: 3x3 product stencil out of LDS ----
#pragma unroll 1
    for (int k = 0; k < PER_THR; ++k) {
        const int idx = tid + THREADS * k;
        const int i = idx >> 6, j = idx & 63;
        float p = 1.0f;
#pragma unroll
        for (int di = -1; di <= 1; ++di) {
            const int ii = i + di;
            const bool rowOK = ((unsigned)ii < (unsigned)WDIM);
#pragma unroll
            for (int dj = -1; dj <= 1; ++dj) {
                const int jj = j + dj;
                const bool ok = rowOK && ((unsigned)jj < (unsigned)WDIM);
                const int a  = ok ? ((ii << 6) + jj) : idx;   // safe addr, unconditional LDS load
                const float v = cb[a];
                p *= ok ? v : cosm;
            }
        }
        out[(size_t)n * PLANE + idx] = p;
    }
}

extern "C" void kernel_launch(void* const* d_in, const int* in_sizes, int n_in,
                              void* d_out, int out_size, void* d_ws, size_t ws_size,
                              hipStream_t stream) {
    (void)in_sizes; (void)n_in; (void)d_ws; (void)ws_size; (void)out_size;
    const float* x = (const float*)d_in[0];     // (4,3,64,64) f32
    float* out = (float*)d_out;                 // (4,3,64,64) f32
    qconv_median_cosprod_kernel<<<dim3(NPLANES), dim3(THREADS), 0, stream>>>(x, out);
}